// MeteoGraphPC_v3_50903952392632
// MI455X (gfx1250) — compile-verified
//
#include <hip/hip_runtime.h>
#include <hip/hip_bf16.h>

typedef float v2f __attribute__((ext_vector_type(2)));
typedef float v4f __attribute__((ext_vector_type(4)));
typedef float v8f __attribute__((ext_vector_type(8)));

// ---------------------------------------------------------------------------
// fp32 WMMA 16x16x4 fragment helpers (CDNA5 ISA 7.12.2 layouts):
//   A 16x4 : lanes 0-15 -> K=0,1 in v0,v1 ; lanes 16-31 -> K=2,3
//   B 4x16 : v0 = rows {0,2}, v1 = rows {1,3} striped across lane halves
//   C 16x16: VGPR r -> M = r (lanes 0-15) / r+8 (lanes 16-31), N = lane%16
// ---------------------------------------------------------------------------

#define LDS_PAD 8          // row pad: 2*(64+8) mod 64 banks == 16 -> conflict-free
#define LDSN    (64 + LDS_PAD)

__device__ __forceinline__ void wmma_store(float* __restrict__ C, int ldc,
                                           int row0, int col0,
                                           const float* __restrict__ bias,
                                           v8f acc, int lane) {
  const int idx   = lane & 15;
  const int rbase = row0 + ((lane >> 4) << 3);
  const float bv  = bias ? bias[col0 + idx] : 0.0f;
#pragma unroll
  for (int r = 0; r < 8; ++r)
    C[(size_t)(rbase + r) * ldc + col0 + idx] = acc[r] + bv;
}

// ---------------------------------------------------------------------------
// Generic GEMM: C(M,N) = A(M,K) @ B(K,N) + bias  (M%16==0, N%64==0, K%4==0)
// Block = 8 waves; each wave owns one 16-row strip and computes 16x64 panels.
// B staged panel-by-panel (Kc=128 x 64 cols) in LDS via float4, shared by all.
// ---------------------------------------------------------------------------
__global__ void gemm_wmma(const float* __restrict__ A, const float* __restrict__ B,
                          const float* __restrict__ bias, float* __restrict__ C,
                          int M, int N, int K) {
  __shared__ float lds[128 * LDSN];           // 36.9 KB
  const int tid   = threadIdx.x;
  const int lane  = tid & 31;
  const int strip = blockIdx.x * 8 + (tid >> 5);
  const bool active = strip < (M >> 4);
  const int row0  = active ? (strip << 4) : 0;
  const int kkoff = (lane >> 4) << 1;          // 0 or 2
  const int idx   = lane & 15;

  for (int cp = 0; cp < N; cp += 64) {
    v8f acc[4] = {};
    for (int kp = 0; kp < K; kp += 128) {
      const int kc = (K - kp < 128) ? (K - kp) : 128;
      // cooperative float4 stage of B[kp..kp+kc)[cp..cp+64) -> LDS (padded rows)
      for (int i = tid; i < kc * 16; i += 256) {
        const int kk = i >> 4, c4 = (i & 15) << 2;
        *(v4f*)&lds[kk * LDSN + c4] =
            *(const v4f*)&B[(size_t)(kp + kk) * N + cp + c4];
      }
      __syncthreads();
      const float* Ab = A + (size_t)row0 * K + kp;
      for (int k0 = 0; k0 < kc; k0 += 4) {
        const int kk = k0 + kkoff;
        v2f a;
        a.x = Ab[(size_t)idx * K + kk];
        a.y = Ab[(size_t)idx * K + kk + 1];
#pragma unroll
        for (int ct = 0; ct < 4; ++ct) {
          v2f b;
          b.x = lds[kk * LDSN + ct * 16 + idx];
          b.y = lds[(kk + 1) * LDSN + ct * 16 + idx];
          acc[ct] = __builtin_amdgcn_wmma_f32_16x16x4_f32(false, a, false, b,
                                                          (short)0, acc[ct], false, false);
        }
      }
      __syncthreads();
    }
    if (active) {
#pragma unroll
      for (int ct = 0; ct < 4; ++ct)
        wmma_store(C, N, row0, cp + ct * 16, bias, acc[ct], lane);
    }
  }
}

// ---------------------------------------------------------------------------
// Dilated 'SAME' conv over T: Y[t] = bias + sum_w X[t+(w-1)*dil] @ W[w]
// X,Y: (T,N,64), W: (3,64,64). All 3 taps staged in LDS once per block.
// ---------------------------------------------------------------------------
__global__ void tcn_conv_wmma(const float* __restrict__ X, const float* __restrict__ W,
                              const float* __restrict__ bias, float* __restrict__ Y,
                              int T, int Nn, int dil) {
  __shared__ float lds[192 * LDSN];           // 55.3 KB: 3 taps x 64 rows
  const int tid   = threadIdx.x;
  const int lane  = tid & 31;
  const int kkoff = (lane >> 4) << 1;
  const int idx   = lane & 15;

  // stage all 3 tap matrices (W is contiguous (3,64,64)) via float4
  for (int i = tid; i < 3 * 64 * 16; i += 256) {
    const int kk = i >> 4, c4 = (i & 15) << 2;
    *(v4f*)&lds[kk * LDSN + c4] = *(const v4f*)&W[(size_t)kk * 64 + c4];
  }
  __syncthreads();

  const int stripsPerT = Nn >> 4;
  const int strip = blockIdx.x * 8 + (tid >> 5);
  if (strip >= T * stripsPerT) return;
  const int t    = strip / stripsPerT;
  const int row0 = (strip % stripsPerT) << 4;

  v8f acc[4] = {};
#pragma unroll
  for (int w = 0; w < 3; ++w) {
    const int ts = t + (w - 1) * dil;
    if (ts < 0 || ts >= T) continue;
    const float* Ab = X + ((size_t)ts * Nn + row0) * 64;
    for (int k0 = 0; k0 < 64; k0 += 4) {
      const int kk = k0 + kkoff;
      v2f a;
      a.x = Ab[(size_t)idx * 64 + kk];
      a.y = Ab[(size_t)idx * 64 + kk + 1];
#pragma unroll
      for (int ct = 0; ct < 4; ++ct) {
        v2f b;
        b.x = lds[(w * 64 + kk) * LDSN + ct * 16 + idx];
        b.y = lds[(w * 64 + kk + 1) * LDSN + ct * 16 + idx];
        acc[ct] = __builtin_amdgcn_wmma_f32_16x16x4_f32(false, a, false, b,
                                                        (short)0, acc[ct], false, false);
      }
    }
  }
#pragma unroll
  for (int ct = 0; ct < 4; ++ct)
    wmma_store(Y + (size_t)t * Nn * 64, 64, row0, ct * 16, bias, acc[ct], lane);
}

// Wave-per-row layernorm over 64 channels; optional residual add & exact GELU.
__global__ void ln64(const float* __restrict__ x, const float* __restrict__ resid,
                     const float* __restrict__ g, const float* __restrict__ beta,
                     float* __restrict__ y, int rows, int gelu) {
  const int wid  = (blockIdx.x * blockDim.x + threadIdx.x) >> 5;
  const int lane = threadIdx.x & 31;
  if (wid >= rows) return;
  const float* xr = x + (size_t)wid * 64;
  float a0 = xr[lane], a1 = xr[lane + 32];
  if (resid) {
    const float* rr = resid + (size_t)wid * 64;
    a0 += rr[lane]; a1 += rr[lane + 32];
  }
  float s = a0 + a1;
#pragma unroll
  for (int m = 16; m > 0; m >>= 1) s += __shfl_xor(s, m, 32);
  const float mean = s * (1.0f / 64.0f);
  const float d0 = a0 - mean, d1 = a1 - mean;
  float v = d0 * d0 + d1 * d1;
#pragma unroll
  for (int m = 16; m > 0; m >>= 1) v += __shfl_xor(v, m, 32);
  const float inv = rsqrtf(v * (1.0f / 64.0f) + 1e-5f);
  float o0 = g[lane] * d0 * inv + beta[lane];
  float o1 = g[lane + 32] * d1 * inv + beta[lane + 32];
  if (gelu) {
    o0 = 0.5f * o0 * (1.0f + erff(o0 * 0.70710678118f));
    o1 = 0.5f * o1 * (1.0f + erff(o1 * 0.70710678118f));
  }
  float* yr = y + (size_t)wid * 64;
  yr[lane] = o0; yr[lane + 32] = o1;
}

// ----------------------- graph helper kernels ------------------------------
__global__ void fill_f32(float* p, int n, float v) {
  int i = blockIdx.x * blockDim.x + threadIdx.x;
  if (i < n) p[i] = v;
}
__global__ void deg_scatter(float* deg, const int* __restrict__ dst, int E) {
  int i = blockIdx.x * blockDim.x + threadIdx.x;
  if (i < E) atomicAdd(&deg[dst[i]], 1.0f);
}
__global__ void rsqrt_ip(float* p, int n) {
  int i = blockIdx.x * blockDim.x + threadIdx.x;
  if (i < n) p[i] = rsqrtf(p[i]);
}
// g = b + xw * dinv[n]^2  (self-loop term); 4 channels per thread
__global__ void gcn_init(float* __restrict__ g, const float* __restrict__ xw,
                         const float* __restrict__ b, const float* __restrict__ dinv, int N) {
  int i = blockIdx.x * blockDim.x + threadIdx.x;
  if (i >= N * 16) return;
  const int n = i >> 4, c4 = (i & 15) << 2;
  const float di = dinv[n];
  const float d2 = di * di;
  const v4f xv = *(const v4f*)&xw[(size_t)n * 64 + c4];
  const v4f bv = *(const v4f*)&b[c4];
  v4f o;
  o.x = bv.x + xv.x * d2; o.y = bv.y + xv.y * d2;
  o.z = bv.z + xv.z * d2; o.w = bv.w + xv.w * d2;
  *(v4f*)&g[(size_t)n * 64 + c4] = o;
}
// 4 channels per thread: one float4 gather of xw + 4 scalar atomics
__global__ void gcn_scatter(float* __restrict__ g, const float* __restrict__ xw,
                            const int* __restrict__ src, const int* __restrict__ dst,
                            const float* __restrict__ dinv, int E) {
  int i = blockIdx.x * blockDim.x + threadIdx.x;
  if (i >= E * 16) return;
  const int e = i >> 4, c4 = (i & 15) << 2;
  const int s = src[e], d = dst[e];
  const float nrm = dinv[s] * dinv[d];
  const v4f xv = *(const v4f*)&xw[(size_t)s * 64 + c4];
  float* gp = &g[(size_t)d * 64 + c4];
  atomicAdd(gp + 0, xv.x * nrm);
  atomicAdd(gp + 1, xv.y * nrm);
  atomicAdd(gp + 2, xv.z * nrm);
  atomicAdd(gp + 3, xv.w * nrm);
}
__global__ void relu_ip(float* p, int n4) {   // n4 = count of float4
  int i = blockIdx.x * blockDim.x + threadIdx.x;
  if (i >= n4) return;
  v4f v = *(v4f*)&p[(size_t)i * 4];
  v.x = fmaxf(v.x, 0.0f); v.y = fmaxf(v.y, 0.0f);
  v.z = fmaxf(v.z, 0.0f); v.w = fmaxf(v.w, 0.0f);
  *(v4f*)&p[(size_t)i * 4] = v;
}

// order-preserving float <-> uint key (for atomicMax segment max)
__device__ __forceinline__ unsigned fkey(float f) {
  unsigned b = __float_as_uint(f);
  return (b & 0x80000000u) ? ~b : (b | 0x80000000u);
}
__device__ __forceinline__ float fdecode(unsigned k) {
  unsigned b = (k & 0x80000000u) ? (k & 0x7FFFFFFFu) : ~k;
  return __uint_as_float(b);
}

// one wave per (edge, head): alpha = dot(q[dst], q[src]+e)/8 ; segment max
__global__ void attn_alpha(const float* __restrict__ q, const int* __restrict__ src,
                           const int* __restrict__ dst, const float* __restrict__ ea,
                           const float* __restrict__ We, float* __restrict__ alpha,
                           unsigned* __restrict__ mkey, int E) {
  const int wid  = (blockIdx.x * blockDim.x + threadIdx.x) >> 5;
  const int lane = threadIdx.x & 31;
  if (wid >= E * 4) return;
  const int e = wid >> 2, h = wid & 3;
  const int s = src[e], d = dst[e];
  const float* qs  = q + (size_t)s * 256 + h * 64;
  const float* qd  = q + (size_t)d * 256 + h * 64;
  const float* eap = ea + (size_t)e * 8;
  float part = 0.0f;
#pragma unroll
  for (int j = 0; j < 2; ++j) {
    const int dd = lane + 32 * j;
    float ev = 0.0f;
#pragma unroll
    for (int k = 0; k < 8; ++k) ev += eap[k] * We[k * 256 + h * 64 + dd];
    part += qd[dd] * (qs[dd] + ev);
  }
#pragma unroll
  for (int m = 16; m > 0; m >>= 1) part += __shfl_xor(part, m, 32);
  if (lane == 0) {
    const float a = part * 0.125f;  // / sqrt(64)
    alpha[wid] = a;
    atomicMax(&mkey[(size_t)d * 4 + h], fkey(a));
  }
}

// w = exp(alpha - max[dst]); denom[dst] += w   (alpha overwritten with w)
__global__ void attn_w(float* __restrict__ alpha, const unsigned* __restrict__ mkey,
                       float* __restrict__ denom, const int* __restrict__ dst, int E) {
  int i = blockIdx.x * blockDim.x + threadIdx.x;
  if (i >= E * 4) return;
  const int e = i >> 2, h = i & 3;
  const int d = dst[e];
  const float m = fdecode(mkey[(size_t)d * 4 + h]);
  const float w = expf(alpha[i] - m);
  alpha[i] = w;
  atomicAdd(&denom[(size_t)d * 4 + h], w);
}

// out[dst] += (q[src]+e) * (w/denom[dst]); one wave per (edge, head)
__global__ void attn_aggr(const float* __restrict__ q, const int* __restrict__ src,
                          const int* __restrict__ dst, const float* __restrict__ ea,
                          const float* __restrict__ We, const float* __restrict__ w,
                          const float* __restrict__ denom, float* __restrict__ aggr, int E) {
  const int wid  = (blockIdx.x * blockDim.x + threadIdx.x) >> 5;
  const int lane = threadIdx.x & 31;
  if (wid >= E * 4) return;
  const int e = wid >> 2, h = wid & 3;
  const int s = src[e], d = dst[e];
  const float a = w[wid] / denom[(size_t)d * 4 + h];
  const float* qs  = q + (size_t)s * 256 + h * 64;
  const float* eap = ea + (size_t)e * 8;
  float* op = aggr + (size_t)d * 256 + h * 64;
#pragma unroll
  for (int j = 0; j < 2; ++j) {
    const int dd = lane + 32 * j;
    float ev = 0.0f;
#pragma unroll
    for (int k = 0; k < 8; ++k) ev += eap[k] * We[k * 256 + h * 64 + dd];
    atomicAdd(&op[dd], (qs[dd] + ev) * a);
  }
}

// out(row,4) = X(row,64) @ head_W(64,4) + head_b
__global__ void head_kernel(const float* __restrict__ X, const float* __restrict__ W,
                            const float* __restrict__ b, float* __restrict__ out, int rows) {
  int i = blockIdx.x * blockDim.x + threadIdx.x;
  if (i >= rows * 4) return;
  const int r = i >> 2, c = i & 3;
  const float* xr = X + (size_t)r * 64;
  float s = b[c];
#pragma unroll
  for (int k = 0; k < 64; ++k) s += xr[k] * W[k * 4 + c];
  out[i] = s;
}

// ---------------------------------------------------------------------------
extern "C" void kernel_launch(void* const* d_in, const int* in_sizes, int n_in,
                              void* d_out, int out_size, void* d_ws, size_t ws_size,
                              hipStream_t stream) {
  (void)in_sizes; (void)n_in; (void)out_size; (void)ws_size;
  const int T = 6, N = 20000, E = 80000, H = 64, HD = 4;
  const int TN = T * N;

  const float* x     = (const float*)d_in[0];
  const int*   eidx  = (const int*)d_in[1];     // (T,2,E)
  const float* eattr = (const float*)d_in[2];   // (T,E,8)
  const float* in_W  = (const float*)d_in[3];
  const float* in_b  = (const float*)d_in[4];
  // tcn layer l: 5+4l (W), 6+4l (b), 7+4l (g), 8+4l (beta)
  // gcn layer l: 21+2l (W), 22+2l (b)
  // attn layer l: 25+6l (Wn), +1 We, +2 Wo, +3 bo, +4 g, +5 beta
  // 37 ln_g, 38 ln_b, 39 head_W, 40 head_b

  float* ws = (float*)d_ws;
  size_t off = 0;
  float*    bufH   = ws + off; off += (size_t)TN * H;       // h / final LN scratch
  float*    bufH2  = ws + off; off += (size_t)TN * H;       // TCN pong / h_out
  float*    bufXW  = ws + off; off += (size_t)N * H;        // xw / Wo tmp
  float*    bufG   = ws + off; off += (size_t)N * H;        // relu(gcn) (residual)
  float*    bufHT  = ws + off; off += (size_t)N * H;        // ht after layer 0
  float*    bufQ   = ws + off; off += (size_t)N * H * HD;   // q (N,4,64)
  float*    bufAgg = ws + off; off += (size_t)N * H * HD;   // attn aggregate
  float*    bufAl  = ws + off; off += (size_t)E * HD;       // alpha / w
  unsigned* bufM   = (unsigned*)(ws + off); off += (size_t)N * HD;  // max keys
  float*    bufDen = ws + off; off += (size_t)N * HD;       // softmax denom
  float*    bufDin = ws + off; off += (size_t)N;            // deg -> rsqrt

  auto gemmBlocks = [](int M) { return ((M / 16) + 7) / 8; };  // strips/8

  // 1) input projection: (TN,16) @ (16,64) + b
  gemm_wmma<<<gemmBlocks(TN), 256, 0, stream>>>(x, in_W, in_b, bufH, TN, H, 16);

  // 2) TCN: 4 dilated conv layers, each followed by LN + exact GELU
  float* cur = bufH; float* oth = bufH2;
  for (int l = 0; l < 4; ++l) {
    const float* Wl = (const float*)d_in[5 + 4 * l];
    const float* bl = (const float*)d_in[6 + 4 * l];
    const float* gl = (const float*)d_in[7 + 4 * l];
    const float* be = (const float*)d_in[8 + 4 * l];
    tcn_conv_wmma<<<(T * (N / 16) + 7) / 8, 256, 0, stream>>>(cur, Wl, bl, oth, T, N, 1 << l);
    ln64<<<(TN + 7) / 8, 256, 0, stream>>>(oth, (const float*)nullptr, gl, be, oth, TN, 1);
    float* tmp = cur; cur = oth; oth = tmp;
  }
  // after 4 layers: cur == bufH holds final h; bufH2 free -> reused as h_out

  // 3) per-timestep graph layers
  for (int t = 0; t < T; ++t) {
    const int*   srcp = eidx + (size_t)t * 2 * E;
    const int*   dstp = srcp + E;
    const float* eap  = eattr + (size_t)t * E * 8;

    fill_f32<<<(N + 255) / 256, 256, 0, stream>>>(bufDin, N, 1.0f);  // self-loops
    deg_scatter<<<(E + 255) / 256, 256, 0, stream>>>(bufDin, dstp, E);
    rsqrt_ip<<<(N + 255) / 256, 256, 0, stream>>>(bufDin, N);

    const float* htin = cur + (size_t)t * N * H;
    for (int l = 0; l < 2; ++l) {
      const float* gW = (const float*)d_in[21 + 2 * l];
      const float* gb = (const float*)d_in[22 + 2 * l];
      const float* Wn = (const float*)d_in[25 + 6 * l];
      const float* We = (const float*)d_in[26 + 6 * l];
      const float* Wo = (const float*)d_in[27 + 6 * l];
      const float* bo = (const float*)d_in[28 + 6 * l];
      const float* ag = (const float*)d_in[29 + 6 * l];
      const float* ab = (const float*)d_in[30 + 6 * l];
      float* htout = (l == 0) ? bufHT : (bufH2 + (size_t)t * N * H);

      // GCN: xw = ht @ W; scatter with sym-norm; + b; relu
      gemm_wmma<<<gemmBlocks(N), 256, 0, stream>>>(htin, gW, (const float*)nullptr,
                                                   bufXW, N, H, H);
      gcn_init<<<(N * 16 + 255) / 256, 256, 0, stream>>>(bufG, bufXW, gb, bufDin, N);
      gcn_scatter<<<(E * 16 + 255) / 256, 256, 0, stream>>>(bufG, bufXW, srcp, dstp,
                                                            bufDin, E);
      relu_ip<<<(N * 16 + 255) / 256, 256, 0, stream>>>(bufG, N * 16);

      // Attention
      gemm_wmma<<<gemmBlocks(N), 256, 0, stream>>>(bufG, Wn, (const float*)nullptr,
                                                   bufQ, N, H * HD, H);
      hipMemsetAsync(bufM, 0, (size_t)N * HD * sizeof(unsigned), stream);
      hipMemsetAsync(bufDen, 0, (size_t)N * HD * sizeof(float), stream);
      attn_alpha<<<(E * HD) / 8, 256, 0, stream>>>(bufQ, srcp, dstp, eap, We, bufAl, bufM, E);
      attn_w<<<(E * HD + 255) / 256, 256, 0, stream>>>(bufAl, bufM, bufDen, dstp, E);
      hipMemsetAsync(bufAgg, 0, (size_t)N * H * HD * sizeof(float), stream);
      attn_aggr<<<(E * HD) / 8, 256, 0, stream>>>(bufQ, srcp, dstp, eap, We, bufAl,
                                                  bufDen, bufAgg, E);
      // out proj + bo, then LN(out + residual g)
      gemm_wmma<<<gemmBlocks(N), 256, 0, stream>>>(bufAgg, Wo, bo, bufXW, N, H, H * HD);
      ln64<<<(N + 7) / 8, 256, 0, stream>>>(bufXW, bufG, ag, ab, htout, N, 0);
      htin = htout;
    }
  }

  // 4) final layernorm + head
  ln64<<<(TN + 7) / 8, 256, 0, stream>>>(bufH2, (const float*)nullptr,
                                         (const float*)d_in[37], (const float*)d_in[38],
                                         bufH, TN, 0);
  head_kernel<<<(TN * 4 + 255) / 256, 256, 0, stream>>>(bufH, (const float*)d_in[39],
                                                        (const float*)d_in[40],
                                                        (float*)d_out, TN);
}